// GIN_24318104830205
// MI455X (gfx1250) — compile-verified
//
#include <hip/hip_runtime.h>
#include <hip/hip_bf16.h>

// GIN forward: 50000 nodes, 800000 edges, 128 -> 64 feat, 3 layers.
// fp32 WMMA (v_wmma_f32_16x16x4_f32) for all GEMMs; BN folded as per-column
// affine (a,b) applied at consumers; edge aggregation via global f32 atomics
// (L2-resident working set on MI455X: ~70MB << 192MB L2).

#define N_NODES 50000
#define N_EDGES 800000
#define NFEAT   128
#define NHID    64
#define NTILES  (N_NODES / 16)   // 3125, exact
#define BN_EPS  1e-5f

typedef float v2f __attribute__((ext_vector_type(2)));
typedef float v8f __attribute__((ext_vector_type(8)));

// ---------------------------------------------------------------------------
// WMMA GEMM: Out[50000 x 64] = f(A[50000 x K] @ B[K x 64])
//   IN_AFFINE: A_eff = A + (Hraw * ascale + bshift)   (per-column affine)
//   HAS_BIAS : add bias[col] to output
//   RELU_OUT : ReLU before store
//   STATS    : accumulate per-column sum / sum^2 of stored values
// One wave computes a 16-row x 64-col strip. Block = 128 threads = 4 waves.
// ---------------------------------------------------------------------------
template <int K, bool HAS_BIAS, bool IN_AFFINE, bool RELU_OUT, bool STATS>
__global__ __launch_bounds__(128)
void gemm_wmma(const float* __restrict__ A,
               const float* __restrict__ Hraw,
               const float* __restrict__ ascale,
               const float* __restrict__ bshift,
               const float* __restrict__ B,
               const float* __restrict__ bias,
               float* __restrict__ Out,
               float* __restrict__ colsum,
               float* __restrict__ colsq) {
    __shared__ float ssum[NHID];
    __shared__ float ssq[NHID];

    const int lane  = threadIdx.x & 31;
    const int wid   = threadIdx.x >> 5;
    const int tile  = blockIdx.x * 4 + wid;
    const int lrow  = lane & 15;
    const int lhalf = lane >> 4;

    if constexpr (STATS) {
        if (threadIdx.x < NHID) { ssum[threadIdx.x] = 0.f; ssq[threadIdx.x] = 0.f; }
        __syncthreads();
    }

    if (tile < NTILES) {  // wave-uniform guard: EXEC stays all-ones for WMMA
        const int r0 = tile * 16;
        v8f acc[4];
        #pragma unroll
        for (int nt = 0; nt < 4; ++nt) acc[nt] = (v8f)0.f;

        const float* arow = A + (size_t)(r0 + lrow) * K;
        const float* hrow = nullptr;
        if constexpr (IN_AFFINE) hrow = Hraw + (size_t)(r0 + lrow) * K;

        for (int k = 0; k < K; k += 4) {
            const int ka = k + 2 * lhalf;           // this lane's K pair
            v2f af;
            af.x = arow[ka];
            af.y = arow[ka + 1];
            if constexpr (IN_AFFINE) {
                af.x += hrow[ka]     * ascale[ka]     + bshift[ka];
                af.y += hrow[ka + 1] * ascale[ka + 1] + bshift[ka + 1];
            }
            #pragma unroll
            for (int nt = 0; nt < 4; ++nt) {
                v2f bf;
                bf.x = B[(size_t)ka * NHID       + nt * 16 + lrow];
                bf.y = B[(size_t)(ka + 1) * NHID + nt * 16 + lrow];
                acc[nt] = __builtin_amdgcn_wmma_f32_16x16x4_f32(
                    false, af, false, bf, (short)0, acc[nt], false, false);
            }
        }

        #pragma unroll
        for (int nt = 0; nt < 4; ++nt) {
            const int col = nt * 16 + lrow;
            float bs = 0.f;
            if constexpr (HAS_BIAS) bs = bias[col];
            float psum = 0.f, psq = 0.f;
            #pragma unroll
            for (int v = 0; v < 8; ++v) {
                float val = acc[nt][v] + bs;
                if constexpr (RELU_OUT) val = fmaxf(val, 0.f);
                Out[(size_t)(r0 + v + 8 * lhalf) * NHID + col] = val;
                if constexpr (STATS) { psum += val; psq += val * val; }
            }
            if constexpr (STATS) {
                atomicAdd(&ssum[col], psum);   // ds_add_f32
                atomicAdd(&ssq[col],  psq);
            }
        }
    }

    if constexpr (STATS) {
        __syncthreads();
        if (threadIdx.x < NHID) {
            atomicAdd(&colsum[threadIdx.x], ssum[threadIdx.x]);
            atomicAdd(&colsq[threadIdx.x],  ssq[threadIdx.x]);
        }
    }
}

// BN: turn (sum, sumsq, gamma, beta) into per-column affine h = h_raw*a + b
__global__ void bnstats_kernel(const float* __restrict__ colsum,
                               const float* __restrict__ colsq,
                               const float* __restrict__ gamma,
                               const float* __restrict__ beta,
                               float* __restrict__ a, float* __restrict__ b,
                               float invN) {
    const int c = threadIdx.x;
    if (c < NHID) {
        const float mu  = colsum[c] * invN;
        const float var = colsq[c] * invN - mu * mu;  // biased variance
        const float s   = gamma[c] * rsqrtf(var + BN_EPS);
        a[c] = s;
        b[c] = beta[c] - mu * s;
    }
}

__global__ void zerof_kernel(float* __restrict__ p, int n) {
    const int i = blockIdx.x * blockDim.x + threadIdx.x;
    if (i < n) p[i] = 0.f;
}

// agg[dst] += a*h_raw[src] + b ; 32 lanes per edge, 2 columns per lane
__global__ void scatter_edges(const int* __restrict__ ei,
                              const float* __restrict__ hraw,
                              const float* __restrict__ a,
                              const float* __restrict__ b,
                              float* __restrict__ agg) {
    const long long gid = (long long)blockIdx.x * blockDim.x + threadIdx.x;
    const int e = (int)(gid >> 5);
    if (e >= N_EDGES) return;
    const int cp  = ((int)gid & 31) * 2;
    const int src = ei[e];
    const int dst = ei[N_EDGES + e];
    const float* hp = hraw + (size_t)src * NHID + cp;
    const float v0 = hp[0] * a[cp]     + b[cp];
    const float v1 = hp[1] * a[cp + 1] + b[cp + 1];
    float* op = agg + (size_t)dst * NHID + cp;
    atomicAdd(op,     v0);   // global_atomic_add_f32
    atomicAdd(op + 1, v1);
}

__global__ void apply_affine(const float* __restrict__ hraw,
                             const float* __restrict__ a,
                             const float* __restrict__ b,
                             float* __restrict__ out, int n) {
    const int i = blockIdx.x * blockDim.x + threadIdx.x;
    if (i < n) {
        const int c = i & (NHID - 1);
        out[i] = hraw[i] * a[c] + b[c];
    }
}

extern "C" void kernel_launch(void* const* d_in, const int* in_sizes, int n_in,
                              void* d_out, int out_size, void* d_ws, size_t ws_size,
                              hipStream_t stream) {
    (void)in_sizes; (void)n_in; (void)out_size; (void)ws_size;
    const float* x      = (const float*)d_in[0];
    const int*   ei     = (const int*)d_in[1];     // edge_index [2, E] (int32)
    const float* W_t    = (const float*)d_in[2];   // [128, 64]
    const float* b_t    = (const float*)d_in[3];
    const float* g_t    = (const float*)d_in[4];
    const float* be_t   = (const float*)d_in[5];
    const float* Ws1    = (const float*)d_in[6];   // [3, 64, 64]
    const float* Ws2    = (const float*)d_in[7];
    const float* gammas = (const float*)d_in[8];
    const float* betas  = (const float*)d_in[9];
    float* out = (float*)d_out;

    const int NE = N_NODES * NHID;                 // 3,200,000
    float* ws     = (float*)d_ws;
    float* hraw   = ws;                            // [N, 64]
    float* agg    = ws + (size_t)NE;               // [N, 64]
    float* tbuf   = ws + (size_t)2 * NE;           // [N, 64]
    float* colsum = ws + (size_t)3 * NE;           // [64]
    float* colsq  = colsum + NHID;                 // [64]
    float* av     = colsum + 2 * NHID;             // [64]
    float* bv     = colsum + 3 * NHID;             // [64]

    const float invN = 1.0f / (float)N_NODES;
    const dim3 gemmGrid((NTILES + 3) / 4);         // 782 blocks x 4 waves
    const dim3 gemmBlk(128);
    const int zeroBlocks = (NE + 255) / 256;
    const int scatBlocks = (int)(((long long)N_EDGES * 32 + 255) / 256);

    // Input transform: h_raw = x @ W_t + b_t, stats on pre-BN values
    zerof_kernel<<<1, 128, 0, stream>>>(colsum, 2 * NHID);
    gemm_wmma<NFEAT, true, false, false, true><<<gemmGrid, gemmBlk, 0, stream>>>(
        x, nullptr, nullptr, nullptr, W_t, b_t, hraw, colsum, colsq);
    bnstats_kernel<<<1, 64, 0, stream>>>(colsum, colsq, g_t, be_t, av, bv, invN);

    for (int i = 0; i < 3; ++i) {
        zerof_kernel<<<zeroBlocks, 256, 0, stream>>>(agg, NE);
        zerof_kernel<<<1, 128, 0, stream>>>(colsum, 2 * NHID);
        // agg[dst] += h[src]   (h = a*h_raw + b folded in)
        scatter_edges<<<scatBlocks, 256, 0, stream>>>(ei, hraw, av, bv, agg);
        // t = relu((agg + h) @ Ws1[i])
        gemm_wmma<NHID, false, true, true, false><<<gemmGrid, gemmBlk, 0, stream>>>(
            agg, hraw, av, bv, Ws1 + (size_t)i * NHID * NHID, nullptr,
            tbuf, nullptr, nullptr);
        // h_raw = relu(t @ Ws2[i]), stats for BN
        gemm_wmma<NHID, false, false, true, true><<<gemmGrid, gemmBlk, 0, stream>>>(
            tbuf, nullptr, nullptr, nullptr, Ws2 + (size_t)i * NHID * NHID, nullptr,
            hraw, colsum, colsq);
        bnstats_kernel<<<1, 64, 0, stream>>>(colsum, colsq,
                                             gammas + i * NHID, betas + i * NHID,
                                             av, bv, invN);
    }

    // final BN application
    apply_affine<<<zeroBlocks, 256, 0, stream>>>(hraw, av, bv, out, NE);
}